// Wasserstein2d_20675972563797
// MI455X (gfx1250) — compile-verified
//
#include <hip/hip_runtime.h>
#include <stdint.h>

#define NT      4096            // time samples per column (sort length)
#define NCOLS   4096            // NTRACES(512) * CHANNELS(8)
#define THREADS 256             // 8 wave32 per block
#define EPT     (NT / THREADS)  // 16 elements per thread

typedef __attribute__((address_space(3))) void as3_void;

// ---- CDNA5 async global->LDS DMA (tracked by ASYNCcnt) ----
__device__ __forceinline__ void async_ld_b32(const float* gsrc, float* ldst) {
#if __has_builtin(__builtin_amdgcn_global_load_async_to_lds_b32)
  // Probe-derived signature: (int* global_src, int* lds_dst, imm offset, imm cpol)
  __builtin_amdgcn_global_load_async_to_lds_b32(
      (int*)gsrc, (int*)ldst, /*offset=*/0, /*cpol=*/0);
#else
  as3_void* lp = (as3_void*)ldst;
  asm volatile("global_load_async_to_lds_b32 %0, %1, off"
               :: "v"(lp), "v"(gsrc) : "memory");
#endif
}

__device__ __forceinline__ void wait_async0() {
#if __has_builtin(__builtin_amdgcn_s_wait_asynccnt)
  __builtin_amdgcn_s_wait_asynccnt(0);
#else
  asm volatile("s_wait_asynccnt 0" ::: "memory");
#endif
}

// One block per (trace, channel) column. Async-DMA the strided column into
// LDS, bitonic-sort pred & obs columns simultaneously, accumulate |diff|.
__global__ void __launch_bounds__(THREADS)
wass_col_sort_kernel(const float* __restrict__ pred,
                     const float* __restrict__ obs,
                     float* __restrict__ partial) {
  __shared__ float sA[NT];
  __shared__ float sB[NT];
  __shared__ float red[THREADS];

  const int col = blockIdx.x;
  const int tid = threadIdx.x;

  // Strided column gather (element stride = NCOLS*4B = 16KB). Each 128B line
  // is shared by 32 adjacent blocks -> L2 (192MB >= dataset) absorbs reuse.
  for (int t = tid; t < NT; t += THREADS) {
    const size_t g = (size_t)t * NCOLS + col;
    async_ld_b32(pred + g, &sA[t]);
    async_ld_b32(obs  + g, &sB[t]);
  }
  wait_async0();
  __syncthreads();

  // Bitonic sort (ascending), both arrays in the same passes.
  for (int k = 2; k <= NT; k <<= 1) {
    // LDS pair-passes for stride j >= 16 (inter-thread partners).
    for (int j = k >> 1; j >= 16; j >>= 1) {
      for (int p = tid; p < NT / 2; p += THREADS) {
        const int i   = ((p & ~(j - 1)) << 1) | (p & (j - 1));
        const int ixj = i | j;
        const bool up = ((i & k) == 0);
        float a = sA[i], b = sA[ixj];
        float lo = fminf(a, b), hi = fmaxf(a, b);
        sA[i]   = up ? lo : hi;
        sA[ixj] = up ? hi : lo;
        float c = sB[i], d = sB[ixj];
        float l2 = fminf(c, d), h2 = fmaxf(c, d);
        sB[i]   = up ? l2 : h2;
        sB[ixj] = up ? h2 : l2;
      }
      __syncthreads();
    }
    // Register-fused phase: j in {8,4,2,1} (clamped by k/2). Each thread owns
    // 16 contiguous elements, so all partners are intra-thread: one LDS
    // load/store round replaces up to 4 pair-passes.
    {
      const int base = tid * EPT;
      float va[EPT], vb[EPT];
      #pragma unroll
      for (int n = 0; n < EPT; ++n) { va[n] = sA[base + n]; vb[n] = sB[base + n]; }
      #pragma unroll
      for (int jj = 8; jj > 0; jj >>= 1) {
        if ((k >> 1) >= jj) {
          #pragma unroll
          for (int n = 0; n < EPT; ++n) {
            if ((n & jj) == 0) {
              const int m = n | jj;
              const bool up = (((base + n) & k) == 0);
              float a = va[n], b = va[m];
              va[n] = up ? fminf(a, b) : fmaxf(a, b);
              va[m] = up ? fmaxf(a, b) : fminf(a, b);
              float c = vb[n], d = vb[m];
              vb[n] = up ? fminf(c, d) : fmaxf(c, d);
              vb[m] = up ? fmaxf(c, d) : fminf(c, d);
            }
          }
        }
      }
      #pragma unroll
      for (int n = 0; n < EPT; ++n) { sA[base + n] = va[n]; sB[base + n] = vb[n]; }
      __syncthreads();
    }
  }

  // Sum |sorted_pred - sorted_obs| for this column (deterministic tree).
  float acc = 0.f;
  for (int t = tid; t < NT; t += THREADS) acc += fabsf(sA[t] - sB[t]);
  red[tid] = acc;
  __syncthreads();
  for (int s = THREADS >> 1; s > 0; s >>= 1) {
    if (tid < s) red[tid] += red[tid + s];
    __syncthreads();
  }
  if (tid == 0) partial[col] = red[0];
}

// Deterministic final reduction of 4096 column partials -> scalar mean.
__global__ void __launch_bounds__(THREADS)
wass_final_reduce(const float* __restrict__ partial, float* __restrict__ out) {
  __shared__ float red[THREADS];
  float acc = 0.f;
  for (int i = threadIdx.x; i < NCOLS; i += THREADS) acc += partial[i];
  red[threadIdx.x] = acc;
  __syncthreads();
  for (int s = THREADS >> 1; s > 0; s >>= 1) {
    if (threadIdx.x < s) red[threadIdx.x] += red[threadIdx.x + s];
    __syncthreads();
  }
  if (threadIdx.x == 0)
    out[0] = red[0] * (1.0f / ((float)NT * (float)NCOLS));  // /2^24 exact
}

extern "C" void kernel_launch(void* const* d_in, const int* in_sizes, int n_in,
                              void* d_out, int out_size, void* d_ws, size_t ws_size,
                              hipStream_t stream) {
  (void)in_sizes; (void)n_in; (void)out_size; (void)ws_size;
  const float* pred = (const float*)d_in[0];   // (NT, NTRACES, CHANNELS) f32
  const float* obs  = (const float*)d_in[1];   // (NT, NTRACES, CHANNELS) f32
  float* partial = (float*)d_ws;               // 4096 floats = 16 KB scratch
  float* out     = (float*)d_out;              // scalar f32

  wass_col_sort_kernel<<<NCOLS, THREADS, 0, stream>>>(pred, obs, partial);
  wass_final_reduce<<<1, THREADS, 0, stream>>>(partial, out);
}